// GATModel_64673617543541
// MI455X (gfx1250) — compile-verified
//
#include <hip/hip_runtime.h>
#include <hip/hip_bf16.h>
#include <math.h>

typedef float v2f __attribute__((ext_vector_type(2)));
typedef float v8f __attribute__((ext_vector_type(8)));

#define NEG_SLOPE 0.2f

// ---------------------------------------------------------------------------
// Float atomic max via sign-aware integer atomics (monotone bit ordering).
// Works with init value -inf; correct for mixed signs.
// ---------------------------------------------------------------------------
__device__ __forceinline__ void atomicMaxF(float* addr, float val) {
  if (val >= 0.0f)
    atomicMax((int*)addr, __float_as_int(val));
  else
    atomicMin((unsigned int*)addr, __float_as_uint(val));
}

// ---------------------------------------------------------------------------
// fp32 WMMA GEMM: Out[nrows x OUTC] = A[nrows x K] @ W[K x OUTC] (+bias)
// One wave computes a 16 x OUTC strip using V_WMMA_F32_16X16X4_F32.
// nrows must be a multiple of 16 (N=100000 = 6250*16).
// ---------------------------------------------------------------------------
template <int K, int OUTC>
__global__ void gemm_wmma(const float* __restrict__ A,
                          const float* __restrict__ W,
                          float* __restrict__ Out,
                          int nrows,
                          const float* __restrict__ bias) {
  const int wave = (blockIdx.x * blockDim.x + threadIdx.x) >> 5;
  const int lane = threadIdx.x & 31;
  if (wave * 16 >= nrows) return;

  constexpr int NT = OUTC / 16;
  v8f acc[NT];
  const v8f zero = {0.f, 0.f, 0.f, 0.f, 0.f, 0.f, 0.f, 0.f};
#pragma unroll
  for (int t = 0; t < NT; ++t) acc[t] = zero;

  // A fragment addressing (16x4 fp32): lanes 0-15 -> K+0/K+1, lanes 16-31 -> K+2/K+3
  const int arow  = wave * 16 + (lane & 15);
  const int khalf = (lane >> 4) * 2;

  for (int k = 0; k < K; k += 4) {
    v2f a;
    a.x = A[(size_t)arow * K + k + khalf];
    a.y = A[(size_t)arow * K + k + khalf + 1];
#pragma unroll
    for (int t = 0; t < NT; ++t) {
      const int col = t * 16 + (lane & 15);
      v2f b;
      b.x = W[(size_t)(k + khalf) * OUTC + col];
      b.y = W[(size_t)(k + khalf + 1) * OUTC + col];
      acc[t] = __builtin_amdgcn_wmma_f32_16x16x4_f32(
          false, a, false, b, (short)0, acc[t], false, false);
    }
  }

  // D layout: lane<16 -> rows v (v=0..7), lane>=16 -> rows v+8; col = (lane&15)
  const int srow = wave * 16 + ((lane >> 4) * 8);
#pragma unroll
  for (int t = 0; t < NT; ++t) {
    const int col = t * 16 + (lane & 15);
    const float badd = bias ? bias[col] : 0.0f;
#pragma unroll
    for (int v = 0; v < 8; ++v) {
      Out[(size_t)(srow + v) * OUTC + col] = acc[t][v] + badd;
    }
  }
}

// ---------------------------------------------------------------------------
// Per-(node,head) attention score dot products:
//   ss[n,h] = sum_c h[n,h,c]*a_src[h,c];  sd likewise.
// ---------------------------------------------------------------------------
template <int H, int C>
__global__ void scores_kernel(const float* __restrict__ h,
                              const float* __restrict__ asrc,
                              const float* __restrict__ adst,
                              float* __restrict__ ss, float* __restrict__ sd,
                              int Nn) {
  const int idx = blockIdx.x * blockDim.x + threadIdx.x;
  if (idx >= Nn * H) return;
  const int n = idx / H;
  const int hh = idx - n * H;
  const float* hp = h + (size_t)n * (H * C) + hh * C;
  const float* as = asrc + hh * C;
  const float* ad = adst + hh * C;
  float s1 = 0.f, s2 = 0.f;
#pragma unroll
  for (int c = 0; c < C; ++c) {
    const float v = hp[c];
    s1 += v * as[c];
    s2 += v * ad[c];
  }
  ss[idx] = s1;
  sd[idx] = s2;
}

__global__ void fill_kernel(float* __restrict__ p, float v, int n) {
  const int i = blockIdx.x * blockDim.x + threadIdx.x;
  if (i < n) p[i] = v;
}

// ---------------------------------------------------------------------------
// Edge pass 1: per-(edge,head) segment max of leaky_relu(ss[src]+sd[dst]).
// Edges e < E_ come from edge_index; e >= E_ are self loops (src=dst=e-E_).
// ---------------------------------------------------------------------------
template <int H>
__global__ void edge_max_kernel(const int* __restrict__ ei, int E_, int Nn,
                                const float* __restrict__ ss,
                                const float* __restrict__ sd,
                                float* __restrict__ m) {
  const long idx = (long)blockIdx.x * blockDim.x + threadIdx.x;
  if (idx >= (long)(E_ + Nn) * H) return;
  const int e = (int)(idx / H);
  const int h = (int)(idx - (long)e * H);
  int s, d;
  if (e < E_) { s = ei[e]; d = ei[E_ + e]; } else { s = d = e - E_; }
  float v = ss[(size_t)s * H + h] + sd[(size_t)d * H + h];
  v = (v >= 0.f) ? v : NEG_SLOPE * v;
  atomicMaxF(&m[(size_t)d * H + h], v);
}

// ---------------------------------------------------------------------------
// Edge pass 2: one wave per edge. w[h] = exp(leaky(e) - m[dst,h]);
// atomicAdd denom[dst,h] and agg[dst, :] += w[h] * feat[src, :].
// Feature width F = H*C (128 for layer1, 64 for layer2); 32 lanes stream F.
// ---------------------------------------------------------------------------
template <int H, int C>
__global__ void edge_agg_kernel(const int* __restrict__ ei, int E_, int Nn,
                                const float* __restrict__ ss,
                                const float* __restrict__ sd,
                                const float* __restrict__ m,
                                const float* __restrict__ feat,
                                float* __restrict__ denom,
                                float* __restrict__ agg) {
  const long wave = ((long)blockIdx.x * blockDim.x + threadIdx.x) >> 5;
  const int lane = threadIdx.x & 31;
  if (wave >= (long)(E_ + Nn)) return;
  int s, d;
  if (wave < E_) { s = ei[wave]; d = ei[E_ + wave]; } else { s = d = (int)wave - E_; }

  float w[H];
#pragma unroll
  for (int h = 0; h < H; ++h) {
    float v = ss[(size_t)s * H + h] + sd[(size_t)d * H + h];
    v = (v >= 0.f) ? v : NEG_SLOPE * v;
    w[h] = __expf(v - m[(size_t)d * H + h]);
  }
  if (lane < H) atomicAdd(&denom[(size_t)d * H + lane], w[lane]);

  constexpr int F = H * C;
#pragma unroll
  for (int it = 0; it < F / 32; ++it) {
    const int c = it * 32 + lane;
    const int h = c / C;
    atomicAdd(&agg[(size_t)d * F + c], feat[(size_t)s * F + c] * w[h]);
  }
}

// ---------------------------------------------------------------------------
// Layer finalizers: normalize by denom, add bias (and ReLU for layer 1).
// ---------------------------------------------------------------------------
__global__ void finalize1_kernel(const float* __restrict__ agg,
                                 const float* __restrict__ den,
                                 const float* __restrict__ bias,
                                 float* __restrict__ out, int Nn) {
  const long idx = (long)blockIdx.x * blockDim.x + threadIdx.x;
  if (idx >= (long)Nn * 128) return;
  const int n = (int)(idx >> 7);
  const int c = (int)(idx & 127);
  float v = agg[idx] / den[(size_t)n * 4 + (c >> 5)] + bias[c];
  out[idx] = (v > 0.f) ? v : 0.f;
}

__global__ void finalize2_kernel(const float* __restrict__ agg,
                                 const float* __restrict__ den,
                                 const float* __restrict__ bias,
                                 float* __restrict__ out, int Nn) {
  const long idx = (long)blockIdx.x * blockDim.x + threadIdx.x;
  if (idx >= (long)Nn * 64) return;
  const int n = (int)(idx >> 6);
  const int c = (int)(idx & 63);
  out[idx] = agg[idx] / den[n] + bias[c];
}

// ---------------------------------------------------------------------------
// Row-wise log_softmax over 64 columns; one wave32 per row, 2 cols per lane.
// ---------------------------------------------------------------------------
__global__ void logsoftmax64_kernel(float* __restrict__ out, int Nn) {
  const long wave = ((long)blockIdx.x * blockDim.x + threadIdx.x) >> 5;
  const int lane = threadIdx.x & 31;
  if (wave >= Nn) return;
  float v0 = out[wave * 64 + lane];
  float v1 = out[wave * 64 + 32 + lane];
  float mx = fmaxf(v0, v1);
#pragma unroll
  for (int off = 16; off > 0; off >>= 1) mx = fmaxf(mx, __shfl_xor(mx, off));
  float se = __expf(v0 - mx) + __expf(v1 - mx);
#pragma unroll
  for (int off = 16; off > 0; off >>= 1) se += __shfl_xor(se, off);
  const float ls = __logf(se);
  out[wave * 64 + lane] = v0 - mx - ls;
  out[wave * 64 + 32 + lane] = v1 - mx - ls;
}

// ---------------------------------------------------------------------------
static inline int cdiv(long a, long b) { return (int)((a + b - 1) / b); }

extern "C" void kernel_launch(void* const* d_in, const int* in_sizes, int n_in,
                              void* d_out, int out_size, void* d_ws, size_t ws_size,
                              hipStream_t stream) {
  const float* x      = (const float*)d_in[0];
  const int*   ei     = (const int*)  d_in[1];   // [2,E] row-major: src row then dst row
  const float* W1     = (const float*)d_in[2];
  const float* a_src1 = (const float*)d_in[3];
  const float* a_dst1 = (const float*)d_in[4];
  const float* b1     = (const float*)d_in[5];
  const float* W2     = (const float*)d_in[6];
  const float* a_src2 = (const float*)d_in[7];
  const float* a_dst2 = (const float*)d_in[8];
  const float* b2     = (const float*)d_in[9];
  const float* fcW    = (const float*)d_in[10];
  const float* fcb    = (const float*)d_in[11];

  const int Nn = in_sizes[0] / 128;   // 100000
  const int Ee = in_sizes[1] / 2;     // 1000000

  // Workspace layout (floats). h2/agg2 alias the h1 region: h1's last reader
  // (edge_agg_kernel<4,32>) is stream-ordered before the layer-2 GEMM writes
  // h2, so the aliasing is safe and cuts peak scratch to ~111 MB.
  float* ws = (float*)d_ws;
  size_t off = 0;
  float* h1   = ws + off; off += (size_t)Nn * 128;  // layer1 features; reused below
  float* agg1 = ws + off; off += (size_t)Nn * 128;  // layer1 aggregate -> x2 (in place)
  float* ss1  = ws + off; off += (size_t)Nn * 4;
  float* sd1  = ws + off; off += (size_t)Nn * 4;
  float* m1   = ws + off; off += (size_t)Nn * 4;
  float* dn1  = ws + off; off += (size_t)Nn * 4;
  float* ss2  = ws + off; off += (size_t)Nn;
  float* sd2  = ws + off; off += (size_t)Nn;
  float* m2   = ws + off; off += (size_t)Nn;
  float* dn2  = ws + off; off += (size_t)Nn;
  float* h2   = h1;                                 // [Nn*64], aliases dead h1
  float* agg2 = h1 + (size_t)Nn * 64;               // [Nn*64], aliases dead h1

  const int BT = 256;
  const long mwaves = (Nn + 15) / 16;
  const long NE = (long)Ee + Nn;      // edges + self loops

  // ---------------- Layer 1 ----------------
  gemm_wmma<128, 128><<<cdiv(mwaves * 32, BT), BT, 0, stream>>>(x, W1, h1, Nn, nullptr);
  scores_kernel<4, 32><<<cdiv((long)Nn * 4, BT), BT, 0, stream>>>(h1, a_src1, a_dst1, ss1, sd1, Nn);
  fill_kernel<<<cdiv((long)Nn * 4, BT), BT, 0, stream>>>(m1, -INFINITY, Nn * 4);
  fill_kernel<<<cdiv((long)Nn * 4, BT), BT, 0, stream>>>(dn1, 0.0f, Nn * 4);
  fill_kernel<<<cdiv((long)Nn * 128, BT), BT, 0, stream>>>(agg1, 0.0f, Nn * 128);
  edge_max_kernel<4><<<cdiv(NE * 4, BT), BT, 0, stream>>>(ei, Ee, Nn, ss1, sd1, m1);
  edge_agg_kernel<4, 32><<<cdiv(NE * 32, BT), BT, 0, stream>>>(ei, Ee, Nn, ss1, sd1, m1, h1, dn1, agg1);
  finalize1_kernel<<<cdiv((long)Nn * 128, BT), BT, 0, stream>>>(agg1, dn1, b1, agg1, Nn);

  // ---------------- Layer 2 ----------------
  gemm_wmma<128, 64><<<cdiv(mwaves * 32, BT), BT, 0, stream>>>(agg1, W2, h2, Nn, nullptr);
  scores_kernel<1, 64><<<cdiv((long)Nn, BT), BT, 0, stream>>>(h2, a_src2, a_dst2, ss2, sd2, Nn);
  fill_kernel<<<cdiv((long)Nn, BT), BT, 0, stream>>>(m2, -INFINITY, Nn);
  fill_kernel<<<cdiv((long)Nn, BT), BT, 0, stream>>>(dn2, 0.0f, Nn);
  fill_kernel<<<cdiv((long)Nn * 64, BT), BT, 0, stream>>>(agg2, 0.0f, Nn * 64);
  edge_max_kernel<1><<<cdiv(NE, BT), BT, 0, stream>>>(ei, Ee, Nn, ss2, sd2, m2);
  edge_agg_kernel<1, 64><<<cdiv(NE * 32, BT), BT, 0, stream>>>(ei, Ee, Nn, ss2, sd2, m2, h2, dn2, agg2);
  finalize2_kernel<<<cdiv((long)Nn * 64, BT), BT, 0, stream>>>(agg2, dn2, b2, agg2, Nn);

  // ---------------- FC + log_softmax ----------------
  gemm_wmma<64, 64><<<cdiv(mwaves * 32, BT), BT, 0, stream>>>(agg2, fcW, (float*)d_out, Nn, fcb);
  logsoftmax64_kernel<<<cdiv((long)Nn * 32, BT), BT, 0, stream>>>((float*)d_out, Nn);
}